// GraphAttention_31671088841316
// MI455X (gfx1250) — compile-verified
//
#include <hip/hip_runtime.h>
#include <math.h>

// GAT layer for MI455X (gfx1250, wave32).
//
// Pipeline (all on `stream`):
//   1. gat_gemm_wmma : act = x @ W^T      via v_wmma_f32_16x16x4_f32
//   2. gat_scores    : s_dst/s_src = act @ a[:32] / a[32:]
//   3. zero agg/den regions of d_out (atomic accumulators, must be 0 each call)
//   4. gat_edges     : 32 lanes per edge; lane = feature. exp(leaky_relu),
//                      NT streaming stores for messages/e, atomics into agg/den.
//   5. gat_finalize  : agg /= where(den==0,1,den); den = where(...)
//
// Cache strategy (192 MB L2, 23.3 TB/s HBM): the streaming outputs
// (messages: 211 MB, e: 6.6 MB) and the read-once edge_index (25.6 MB) use
// non-temporal hints so the reused working set (act 6.4 MB + scores + agg/den
// atomic targets) stays L2-resident.

typedef __attribute__((ext_vector_type(2))) float v2f;
typedef __attribute__((ext_vector_type(8))) float v8f;

#define IN_F  128
#define OUT_F 32

// ---------------------------------------------------------------------------
// 1. act = x @ W^T  with fp32 WMMA (16x16x4). One wave per 16-row M tile,
//    two 16-wide N tiles (OUT_F = 32). K loop: 128/4 = 32 steps.
//    A layout (16x4 f32, MxK):  a[j] = A[lane%16][k + j + 2*(lane/16)]
//    B layout (4x16 f32, KxN):  b[j] = B[k + j + 2*(lane/16)][lane%16]
//    C/D layout (16x16 f32):    c[v] = D[v + 8*(lane/16)][lane%16]
// ---------------------------------------------------------------------------
__global__ __launch_bounds__(32) void gat_gemm_wmma(
    const float* __restrict__ x,   // [n_nodes, 128]
    const float* __restrict__ W,   // [32, 128]
    float* __restrict__ act,       // [n_nodes, 32]
    int n_nodes)
{
    const int tile = blockIdx.x;           // 16-row tile index
    const int lane = threadIdx.x;          // 0..31
    const int m    = lane & 15;
    const int g    = lane >> 4;            // lane group 0/1
    const int row  = tile * 16 + m;
    const int rowc = row < n_nodes ? row : n_nodes - 1;   // clamp (EXEC must stay full for WMMA)

    const float* __restrict__ xrow = x + (size_t)rowc * IN_F;
    const float* __restrict__ w0   = W + (size_t)m * IN_F;            // W[n]      (n = lane%16)
    const float* __restrict__ w1   = W + (size_t)(m + 16) * IN_F;     // W[n + 16]

    v8f c0 = {};
    v8f c1 = {};

    #pragma unroll
    for (int k = 0; k < IN_F; k += 4) {
        const int kk = k + 2 * g;
        v2f a;  a.x  = xrow[kk];  a.y  = xrow[kk + 1];
        v2f b0; b0.x = w0[kk];    b0.y = w0[kk + 1];
        v2f b1; b1.x = w1[kk];    b1.y = w1[kk + 1];
        c0 = __builtin_amdgcn_wmma_f32_16x16x4_f32(false, a, false, b0, (short)0, c0, false, false);
        c1 = __builtin_amdgcn_wmma_f32_16x16x4_f32(false, a, false, b1, (short)0, c1, false, false);
    }

    #pragma unroll
    for (int v = 0; v < 8; ++v) {
        const int r = tile * 16 + v + 8 * g;
        if (r < n_nodes) {
            act[(size_t)r * OUT_F + m]      = c0[v];
            act[(size_t)r * OUT_F + m + 16] = c1[v];
        }
    }
}

// ---------------------------------------------------------------------------
// 2. Per-node attention score halves.
// ---------------------------------------------------------------------------
__global__ void gat_scores(const float* __restrict__ act,
                           const float* __restrict__ a,
                           float* __restrict__ s_dst,
                           float* __restrict__ s_src,
                           int n_nodes)
{
    const int i = blockIdx.x * blockDim.x + threadIdx.x;
    if (i >= n_nodes) return;
    const float* r = act + (size_t)i * OUT_F;
    float sd = 0.f, ss = 0.f;
    #pragma unroll
    for (int f = 0; f < OUT_F; ++f) {
        const float v = r[f];
        sd = fmaf(v, a[f], sd);
        ss = fmaf(v, a[OUT_F + f], ss);
    }
    s_dst[i] = sd;
    s_src[i] = ss;
}

// ---------------------------------------------------------------------------
// 3. Zero-fill (atomic accumulators in d_out).
// ---------------------------------------------------------------------------
__global__ void gat_zero(float* __restrict__ p, int n)
{
    const int i = blockIdx.x * blockDim.x + threadIdx.x;
    if (i < n) p[i] = 0.f;
}

// ---------------------------------------------------------------------------
// 4. Edge pass. 32 contiguous lanes share one edge; lane index = feature.
//    Coalesced 128B cached gather of act[src]; NT 128B streaming store of
//    messages; one global_atomic_add_f32 per lane into agg[dst]; lane 0
//    handles the edge weight (NT store) + den atomic. Self-loops at eid >= E.
// ---------------------------------------------------------------------------
__global__ void gat_edges(const long long* __restrict__ ei,   // [2, n_edges] int64
                          const float* __restrict__ act,
                          const float* __restrict__ s_dst,
                          const float* __restrict__ s_src,
                          float* __restrict__ e_out,          // [n_edges + n_nodes]
                          float* __restrict__ msg_out,        // [n_edges + n_nodes, 32]
                          float* __restrict__ agg,            // [n_nodes, 32] accum
                          float* __restrict__ den,            // [n_nodes]     accum
                          int n_edges, int n_nodes)
{
    const long long gtid  = (long long)blockIdx.x * blockDim.x + threadIdx.x;
    const long long eid   = gtid >> 5;
    const int       lane  = (int)(gtid & 31);
    const long long total = (long long)n_edges + n_nodes;
    if (eid >= total) return;

    int src, dst;
    if (eid < n_edges) {
        // read-once index stream: non-temporal loads
        src = (int)__builtin_nontemporal_load(&ei[eid]);
        dst = (int)__builtin_nontemporal_load(&ei[(long long)n_edges + eid]);
    } else {
        src = dst = (int)(eid - n_edges);   // self loop
    }

    const float logit = s_dst[dst] + s_src[src];
    const float lr    = logit > 0.f ? logit : 0.01f * logit;   // leaky_relu(0.01)
    const float e     = expf(lr);

    const float m = act[(size_t)src * OUT_F + lane];            // cached gather (L2-resident)
    __builtin_nontemporal_store(m, &msg_out[eid * OUT_F + lane]);  // write-once stream
    atomicAdd(&agg[(size_t)dst * OUT_F + lane], m * e);
    if (lane == 0) {
        __builtin_nontemporal_store(e, &e_out[eid]);            // write-once stream
        atomicAdd(&den[dst], e);
    }
}

// ---------------------------------------------------------------------------
// 5. Finalize: softmax normalization + den "where" rewrite.
// ---------------------------------------------------------------------------
__global__ void gat_finalize(float* __restrict__ agg,
                             float* __restrict__ den,
                             int n_nodes)
{
    const int i    = blockIdx.x * blockDim.x + threadIdx.x;
    const int node = i >> 5;
    const int lane = i & 31;
    if (node >= n_nodes) return;
    const float d  = den[node];
    const float dd = (d == 0.f) ? 1.f : d;
    agg[(size_t)node * OUT_F + lane] /= dd;
    if (lane == 0) den[node] = dd;
}

// ---------------------------------------------------------------------------
extern "C" void kernel_launch(void* const* d_in, const int* in_sizes, int n_in,
                              void* d_out, int out_size, void* d_ws, size_t ws_size,
                              hipStream_t stream)
{
    const float*     x  = (const float*)d_in[0];       // [N, 128]
    const long long* ei = (const long long*)d_in[1];   // [2, E] int64
    const float*     W  = (const float*)d_in[2];       // [32, 128]
    const float*     a  = (const float*)d_in[3];       // [64]

    const int n_nodes = in_sizes[0] / IN_F;
    const int n_edges = in_sizes[1] / 2;
    const long long total_e = (long long)n_edges + n_nodes;

    // d_out layout (tuple flattened in return order):
    //   agg [N*32] | edge_weights [E+N] | softmax_weights [N] | messages [(E+N)*32]
    float* out   = (float*)d_out;
    float* agg   = out;
    float* e_out = agg + (size_t)n_nodes * OUT_F;
    float* den   = e_out + total_e;
    float* msg   = den + n_nodes;

    // d_ws layout: act [N*32] | s_dst [N] | s_src [N]
    float* act   = (float*)d_ws;
    float* s_dst = act + (size_t)n_nodes * OUT_F;
    float* s_src = s_dst + n_nodes;

    // 1. act = x @ W^T (WMMA)
    const int m_tiles = (n_nodes + 15) / 16;
    gat_gemm_wmma<<<m_tiles, 32, 0, stream>>>(x, W, act, n_nodes);

    // 2. attention score halves
    gat_scores<<<(n_nodes + 255) / 256, 256, 0, stream>>>(act, a, s_dst, s_src, n_nodes);

    // 3. zero atomic accumulators (regions of d_out)
    const int agg_n = n_nodes * OUT_F;
    gat_zero<<<(agg_n + 255) / 256, 256, 0, stream>>>(agg, agg_n);
    gat_zero<<<(n_nodes + 255) / 256, 256, 0, stream>>>(den, n_nodes);

    // 4. edge scatter pass: 32 lanes per edge
    const long long threads = total_e * 32;
    const int blocks = (int)((threads + 255) / 256);
    gat_edges<<<blocks, 256, 0, stream>>>(ei, act, s_dst, s_src,
                                          e_out, msg, agg, den,
                                          n_edges, n_nodes);

    // 5. finalize normalization
    const long long fthreads = (long long)n_nodes * 32;
    gat_finalize<<<(int)((fthreads + 255) / 256), 256, 0, stream>>>(agg, den, n_nodes);
}